// Differentiating1D_47021301957043
// MI455X (gfx1250) — compile-verified
//
#include <hip/hip_runtime.h>
#include <cstdint>

// Differentiating1D on x:(16, 8192, 512) f32, channels_last.
// y[b,t,:] = x[b,t+1,:] - x[b,t,:]  for t < L-1;  y[b,L-1,:] = x[b,L-1,:] - x[b,L-2,:]
//
// Pure streaming op: 512 MiB min HBM traffic -> ~23 us floor at 23.3 TB/s.
// WMMA is deliberately NOT used (banded-matmul view costs 64x the FLOPs for
// zero byte savings). Instead we use the CDNA5 async data-movement path:
//   global_load_async_to_lds_b128  (ASYNCcnt)  -> stage tile rows into LDS
//   s_wait_asynccnt 0 + workgroup barrier      -> cross-wave visibility
//   ds_load_b128 from LDS, NT b128 stores      -> compute + drain
// so the 2x temporal reuse of every input row lives in LDS, and output
// stores don't pollute L2.

typedef float __attribute__((ext_vector_type(4))) v4f;   // native vector: OK for NT builtins

#define B_      16
#define L_      8192
#define F4_     128      // 512 floats per row = 128 float4
#define TR_     16       // output timesteps per block
#define CHUNKS_ (L_ / TR_)      // 512 chunks per batch
#define THREADS 256

__global__ __launch_bounds__(THREADS) void
diff1d_async_kernel(const v4f* __restrict__ x, v4f* __restrict__ y) {
    // (TR_+1) rows * 2048 B = 34,816 B of LDS (well under 320 KB/WGP)
    __shared__ v4f tile[(TR_ + 1) * F4_];

    const int  blk       = blockIdx.x;
    const int  chunk     = blk & (CHUNKS_ - 1);   // chunk within batch
    const int  b         = blk >> 9;              // batch id (CHUNKS_ = 512)
    const int  t0        = chunk * TR_;
    const bool lastChunk = (chunk == CHUNKS_ - 1);
    const int  tid       = threadIdx.x;

    // float4 index of the tile's first element in the global tensor
    const long long rowBase = ((long long)b * L_ + t0) * F4_;
    const v4f* __restrict__ src = x + rowBase;

    // Interior chunks stage TR_+1 rows (need t0+TR_ for the forward diff of
    // row t0+TR_-1). The final chunk of each batch stages only TR_ rows; its
    // last row reuses the previous row instead.
    const int nIn4 = (lastChunk ? TR_ : TR_ + 1) * F4_;   // 2048 or 2176

    // ---- Async global -> LDS staging (gfx1250 ASYNCcnt path) ----
    // Per-lane: copies 16 bytes from global address gaddr to LDS address laddr. Waves are entirely
    // active/inactive per loop trip (nIn4 is a multiple of 128), no intra-wave
    // divergence.
    for (int j = tid; j < nIn4; j += THREADS) {
        uint64_t gaddr = (uint64_t)(uintptr_t)(src + j);
        uint32_t laddr = (uint32_t)(uintptr_t)(&tile[j]);
        asm volatile("global_load_async_to_lds_b128 %0, %1, off"
                     :: "v"(laddr), "v"(gaddr)
                     : "memory");
    }
    // Each wave drains its own async counter, then the workgroup barrier
    // publishes the LDS tile to all 8 waves.
    asm volatile("s_wait_asynccnt 0" ::: "memory");
    __syncthreads();

    // ---- Compute diffs from LDS, stream out with NT stores ----
    v4f* __restrict__ dst = y + rowBase;
    #pragma unroll
    for (int k = 0; k < (TR_ * F4_) / THREADS; ++k) {   // 8 iterations
        const int j  = tid + k * THREADS;               // 0..2047
        const int tl = j >> 7;                          // local row 0..15
        const int gT = t0 + tl;                         // global timestep
        // Branchless edge: for t == L-1 shift the read window back one row so
        // y = x[L-1] - x[L-2]. tl (hence jj) is wave-uniform.
        const int jj = (gT == L_ - 1) ? (j - F4_) : j;

        const v4f nxt = tile[jj + F4_];                 // ds_load_b128
        const v4f cur = tile[jj];                       // ds_load_b128

        const v4f r = nxt - cur;                        // 4x v_sub_f32

        // Output is write-once/never-read: non-temporal store keeps L2 for
        // the input stream.
        __builtin_nontemporal_store(r, &dst[j]);
    }
}

extern "C" void kernel_launch(void* const* d_in, const int* in_sizes, int n_in,
                              void* d_out, int out_size, void* d_ws, size_t ws_size,
                              hipStream_t stream) {
    (void)in_sizes; (void)n_in; (void)d_ws; (void)ws_size; (void)out_size;

    const v4f* x = (const v4f*)d_in[0];
    v4f*       y = (v4f*)d_out;

    // 16 batches * 512 chunks = 8192 blocks, 256 threads (8 wave32s) each.
    dim3 grid(B_ * CHUNKS_);
    dim3 block(THREADS);
    diff1d_async_kernel<<<grid, block, 0, stream>>>(x, y);
}